// GCN_36790689858168
// MI455X (gfx1250) — compile-verified
//
#include <hip/hip_runtime.h>

typedef __attribute__((ext_vector_type(2))) float v2f;
typedef __attribute__((ext_vector_type(8))) float v8f;

#define N_IN   128
#define N_HID  256
#define N_HID2 128
#define N_CLS  2

// ---------------- degree ----------------
__global__ void deg_kernel(const int* __restrict__ col, float* __restrict__ deg, int E) {
    int e = blockIdx.x * blockDim.x + threadIdx.x;
    if (e < E) atomicAdd(&deg[col[e]], 1.0f);
}

__global__ void deginv_kernel(const float* __restrict__ deg, float* __restrict__ dinv, int n) {
    int i = blockIdx.x * blockDim.x + threadIdx.x;
    if (i < n) dinv[i] = 1.0f / fmaxf(deg[i], 1.0f);
}

// ---------------- scatter-add aggregation ----------------
// one thread per (edge, feature); F is a power of two
template<int F>
__global__ void scatter_kernel(const float* __restrict__ src,
                               const int* __restrict__ row,
                               const int* __restrict__ col,
                               float* __restrict__ agg,
                               long long total) {
    long long tid = (long long)blockIdx.x * blockDim.x + threadIdx.x;
    if (tid >= total) return;
    int e = (int)(tid >> (__builtin_ctz(F)));
    int f = (int)(tid & (F - 1));
    int r = row[e], c = col[e];
    atomicAdd(&agg[(size_t)c * F + f], src[(size_t)r * F + f]);
}

// ---------------- fused GEMM: out = act(dinv[m] * (A @ W^T) + bias) ----------------
// A: [M, K] row-major, W: [NF, K] row-major (torch Linear layout), out: [M, NF]
// One wave32 computes a 16x16 tile with V_WMMA_F32_16X16X4_F32.
template<int K, int NF, bool RELU>
__global__ __launch_bounds__(256)
void gemm_wmma(const float* __restrict__ A, const float* __restrict__ W,
               const float* __restrict__ bias, const float* __restrict__ dinv,
               float* __restrict__ out, int M) {
    const int lane  = threadIdx.x;        // 0..31
    const int waveId = threadIdx.y;       // 0..7
    const int tilesN = NF / 16;
    const int tilesM = (M + 15) / 16;
    int tile = blockIdx.x * blockDim.y + waveId;
    if (tile >= tilesM * tilesN) return;  // wave-uniform predicate (EXEC stays all-1)
    const int tm = tile / tilesN;
    const int tn = tile % tilesN;

    const int half = lane >> 4;           // 0: K+{0,1}, 1: K+{2,3}
    const int l16  = lane & 15;

    int arow = tm * 16 + l16;
    if (arow >= M) arow = M - 1;          // clamp loads, guard stores later
    const float* __restrict__ aptr = A + (size_t)arow * K;
    const float* __restrict__ wptr = W + (size_t)(tn * 16 + l16) * K;

    v8f acc = {};
    #pragma unroll
    for (int k = 0; k < K; k += 4) {
        const int kk = k + 2 * half;
        v2f a = *(const v2f*)(aptr + kk);  // A[m][kk], A[m][kk+1]
        v2f b = *(const v2f*)(wptr + kk);  // W[n][kk], W[n][kk+1]  == B^T rows
        acc = __builtin_amdgcn_wmma_f32_16x16x4_f32(
            /*neg_a=*/false, a, /*neg_b=*/false, b,
            /*c_mod=*/(short)0, acc, /*reuse_a=*/false, /*reuse_b=*/false);
    }

    const int n  = tn * 16 + l16;
    const float bv = bias[n];
    #pragma unroll
    for (int r = 0; r < 8; ++r) {
        int m = tm * 16 + half * 8 + r;   // D layout: VGPR r -> M = r + 8*(lane>=16)
        if (m < M) {
            float v = acc[r] * dinv[m] + bv;
            if (RELU) v = fmaxf(v, 0.0f);
            out[(size_t)m * NF + n] = v;
        }
    }
}

// ---------------- head: out = h @ W3^T + b3 ----------------
__global__ void head_kernel(const float* __restrict__ h, const float* __restrict__ W3,
                            const float* __restrict__ b3, float* __restrict__ out, int M) {
    int tid = blockIdx.x * blockDim.x + threadIdx.x;
    if (tid >= M * N_CLS) return;
    int m = tid / N_CLS, c = tid % N_CLS;
    const float* __restrict__ hm = h  + (size_t)m * N_HID2;
    const float* __restrict__ wc = W3 + (size_t)c * N_HID2;
    float s = b3[c];
    #pragma unroll 8
    for (int k = 0; k < N_HID2; ++k) s += hm[k] * wc[k];
    out[tid] = s;
}

extern "C" void kernel_launch(void* const* d_in, const int* in_sizes, int n_in,
                              void* d_out, int out_size, void* d_ws, size_t ws_size,
                              hipStream_t stream) {
    const float* x  = (const float*)d_in[0];
    const int*   ei = (const int*)  d_in[1];   // [2, E] int
    const float* W1 = (const float*)d_in[3];
    const float* b1 = (const float*)d_in[4];
    const float* W2 = (const float*)d_in[5];
    const float* b2 = (const float*)d_in[6];
    const float* W3 = (const float*)d_in[7];
    const float* b3 = (const float*)d_in[8];
    float* out = (float*)d_out;

    const int M = in_sizes[0] / N_IN;
    const int E = in_sizes[1] / 2;
    const int* row = ei;
    const int* col = ei + E;

    float* ws = (float*)d_ws;
    size_t off = 0;
    float* deg  = ws + off; off += (size_t)M;
    float* dinv = ws + off; off += (size_t)M;
    float* agg1 = ws + off; off += (size_t)M * N_IN;
    float* h1   = ws + off; off += (size_t)M * N_HID;
    float* agg2 = ws + off; off += (size_t)M * N_HID;
    float* h2   = ws + off; off += (size_t)M * N_HID2;
    (void)ws_size;

    hipMemsetAsync(deg,  0, sizeof(float) * (size_t)M, stream);
    hipMemsetAsync(agg1, 0, sizeof(float) * (size_t)M * N_IN, stream);
    hipMemsetAsync(agg2, 0, sizeof(float) * (size_t)M * N_HID, stream);

    deg_kernel<<<(E + 255) / 256, 256, 0, stream>>>(col, deg, E);
    deginv_kernel<<<(M + 255) / 256, 256, 0, stream>>>(deg, dinv, M);

    // layer 1: aggregate x -> agg1, then h1 = relu(dinv * agg1 @ W1^T + b1)
    long long tot1 = (long long)E * N_IN;
    scatter_kernel<N_IN><<<(unsigned)((tot1 + 255) / 256), 256, 0, stream>>>(x, row, col, agg1, tot1);
    int tiles1 = ((M + 15) / 16) * (N_HID / 16);
    gemm_wmma<N_IN, N_HID, true>
        <<<(tiles1 + 7) / 8, dim3(32, 8), 0, stream>>>(agg1, W1, b1, dinv, h1, M);

    // layer 2: aggregate h1 -> agg2, then h2 = relu(dinv * agg2 @ W2^T + b2)
    long long tot2 = (long long)E * N_HID;
    scatter_kernel<N_HID><<<(unsigned)((tot2 + 255) / 256), 256, 0, stream>>>(h1, row, col, agg2, tot2);
    int tiles2 = ((M + 15) / 16) * (N_HID2 / 16);
    gemm_wmma<N_HID, N_HID2, true>
        <<<(tiles2 + 7) / 8, dim3(32, 8), 0, stream>>>(agg2, W2, b2, dinv, h2, M);

    // head
    head_kernel<<<(M * N_CLS + 255) / 256, 256, 0, stream>>>(h2, W3, b3, out, M);
}